// WunschLineMatcher_8383776161897
// MI455X (gfx1250) — compile-verified
//
#include <hip/hip_runtime.h>
#include <math.h>

// ---------------------------------------------------------------------------
// WunschLineMatcher for MI455X (gfx1250, wave32).
// Core GEMM (5120x5120x128 fp32) runs on V_WMMA_F32_16X16X4_F32 with the
// 10x10-sample score blocks reduced in LDS (never hits HBM).
// ---------------------------------------------------------------------------

#define NLINES 512
#define S      10           // samples per line
#define NPTS   (NLINES * S) // 5120
#define DCH    128          // descriptor channels
#define HC     128
#define WC     128
#define GAPC   0.1f

typedef __attribute__((ext_vector_type(2))) float v2f;
typedef __attribute__((ext_vector_type(8))) float v8f;

// --------------------------------------------------------------------------
// Kernel 1: sample points along each segment (both sides, 1024 segments).
// --------------------------------------------------------------------------
__global__ void sample_pts_kernel(const float* __restrict__ seg1,
                                  const float* __restrict__ seg2,
                                  float2* __restrict__ ptsA, float2* __restrict__ ptsB,
                                  int* __restrict__ valA, int* __restrict__ valB) {
    int idx = blockIdx.x * blockDim.x + threadIdx.x;
    if (idx >= 2 * NLINES) return;
    int side = idx >> 9;
    int li   = idx & (NLINES - 1);
    const float* s = (side ? seg2 : seg1) + (size_t)li * 4;
    float2* pts = side ? ptsB : ptsA;
    int*    val = side ? valB : valA;

    float ay = s[0], ax = s[1], by = s[2], bx = s[3];
    float dy = ay - by, dx = ax - bx;
    float len = sqrtf(dy * dy + dx * dx);
    float nf = floorf(len * (1.0f / 8.0f));          // MIN_DIST_PTS = 8
    nf = fminf(fmaxf(nf, 2.0f), 10.0f);              // clip to [2, NUM_SAMPLES]
    float ivy = (by - ay) / (nf - 1.0f);
    float ivx = (bx - ax) / (nf - 1.0f);
    for (int t = 0; t < S; ++t) {
        bool v = ((float)t < nf);
        float2 p;
        p.x = v ? (ay + (float)t * ivy) : 0.0f;      // y
        p.y = v ? (ax + (float)t * ivx) : 0.0f;      // x
        pts[li * S + t] = p;
        val[li * S + t] = v ? 1 : 0;
    }
}

// --------------------------------------------------------------------------
// Kernel 2: bilinear grid-sample + L2 normalize. One block (128 threads = one
// channel each) per point; 10240 points total.
// --------------------------------------------------------------------------
__global__ void grid_sample_kernel(const float* __restrict__ desc1,
                                   const float* __restrict__ desc2,
                                   const float2* __restrict__ ptsA,
                                   const float2* __restrict__ ptsB,
                                   float* __restrict__ o1, float* __restrict__ o2) {
    int p = blockIdx.x;
    int c = threadIdx.x;
    bool side = (p >= NPTS);
    int  lp   = side ? (p - NPTS) : p;
    const float* desc = side ? desc2 : desc1;
    float2 pt = (side ? ptsB : ptsA)[lp];
    float* out = side ? o2 : o1;

    // img_size = (Hc*4, Wc*4) = (512, 512)
    float gx = 2.0f * pt.y / 511.0f - 1.0f;
    float gy = 2.0f * pt.x / 511.0f - 1.0f;
    float ix = ((gx + 1.0f) * (float)WC - 1.0f) * 0.5f;
    float iy = ((gy + 1.0f) * (float)HC - 1.0f) * 0.5f;
    float x0 = floorf(ix), y0 = floorf(iy);
    float wx1 = ix - x0, wy1 = iy - y0;
    int x0i = (int)x0, y0i = (int)y0;

    auto g = [&](int yi, int xi) -> float {
        bool inb = (xi >= 0) && (xi < WC) && (yi >= 0) && (yi < HC);
        int yc = min(max(yi, 0), HC - 1);
        int xc = min(max(xi, 0), WC - 1);
        float vv = desc[(size_t)c * (HC * WC) + yc * WC + xc];
        return inb ? vv : 0.0f;
    };
    float v = g(y0i, x0i)       * (1.0f - wy1) * (1.0f - wx1)
            + g(y0i, x0i + 1)   * (1.0f - wy1) * wx1
            + g(y0i + 1, x0i)   * wy1 * (1.0f - wx1)
            + g(y0i + 1, x0i+1) * wy1 * wx1;

    __shared__ float red[DCH];
    red[c] = v * v;
    __syncthreads();
    for (int st = DCH / 2; st > 0; st >>= 1) {
        if (c < st) red[c] += red[c + st];
        __syncthreads();
    }
    float nrm = sqrtf(red[0]);
    out[(size_t)lp * DCH + c] = v / fmaxf(nrm, 1e-12f);
}

// --------------------------------------------------------------------------
// Kernel 3: WMMA GEMM (f32 16x16x4) + fused line_scores reduction.
// Block (bx,by): 16x16 line pairs = 160x160 score block = 10x10 WMMA tiles.
// Scores live only in registers + LDS; output is line_scores[512][512].
// --------------------------------------------------------------------------
__global__ __launch_bounds__(256)
void gemm_linescores_kernel(const float* __restrict__ dA, const float* __restrict__ dB,
                            const int* __restrict__ valA, const int* __restrict__ valB,
                            float* __restrict__ lineScores) {
    extern __shared__ float sS[];            // 160*161 floats + 320 ints
    int* sVA = (int*)(sS + 160 * 161);       // 160 row-valid flags
    int* sVB = sVA + 160;                    // 160 col-valid flags

    const int R0 = blockIdx.x * 160;         // global sample-row base (set1)
    const int C0 = blockIdx.y * 160;         // global sample-col base (set2)
    const int tid = threadIdx.x;

    for (int i = tid; i < 160; i += 256) { sVA[i] = valA[R0 + i]; sVB[i] = valB[C0 + i]; }
    __syncthreads();

    const int w    = tid >> 5;
    const int lane = tid & 31;
    const int r    = lane & 15;              // M for A-frag, N for B-frag
    const int koff = (lane >> 4) << 1;       // K sub-offset per lane half

    for (int t = w; t < 100; t += 8) {       // 10x10 tiles, 8 waves
        int ti = t / 10, tj = t % 10;
        const float* Ap = dA + (size_t)(R0 + ti * 16 + r) * DCH + koff;
        const float* Bp = dB + (size_t)(C0 + tj * 16 + r) * DCH + koff;
        v8f acc = {0.f, 0.f, 0.f, 0.f, 0.f, 0.f, 0.f, 0.f};
        #pragma unroll
        for (int k = 0; k < DCH; k += 4) {
            v2f a = *(const v2f*)(Ap + k);
            v2f b = *(const v2f*)(Bp + k);
            acc = __builtin_amdgcn_wmma_f32_16x16x4_f32(
                false, a, false, b, (short)0, acc, false, false);
        }
        // C layout: VGPR v -> M = v + 8*(lane>=16), N = lane%16
        const int n   = lane & 15;
        const int mh  = (lane >> 4) << 3;
        const int col = tj * 16 + n;
        const int cv  = sVB[col];
        #pragma unroll
        for (int v = 0; v < 8; ++v) {
            int row = ti * 16 + mh + v;
            int ok = sVA[row] & cv;
            sS[row * 161 + col] = ok ? acc[v] : -1.0f;
        }
    }
    __syncthreads();

    // One thread per line pair (16x16 = 256): masked mean of row/col maxima.
    const int il = tid >> 4, jl = tid & 15;
    float s1 = 0.f, s2 = 0.f; int c1 = 0, c2 = 0;
    #pragma unroll
    for (int q = 0; q < S; ++q) {
        float m1 = -1e30f, m2 = -1e30f;
        int rowBase = (il * S + q) * 161 + jl * S;
        #pragma unroll
        for (int cc = 0; cc < S; ++cc) m1 = fmaxf(m1, sS[rowBase + cc]);
        #pragma unroll
        for (int rr = 0; rr < S; ++rr) m2 = fmaxf(m2, sS[(il * S + rr) * 161 + jl * S + q]);
        if (m1 != -1.0f) { s1 += m1; ++c1; }
        if (m2 != -1.0f) { s2 += m2; ++c2; }
    }
    lineScores[(size_t)(blockIdx.x * 16 + il) * NLINES + (blockIdx.y * 16 + jl)] =
        0.5f * (s1 / (float)c1 + s2 / (float)c2);
}

// --------------------------------------------------------------------------
// Kernel 4: per-line top-10 + Needleman-Wunsch over 20 candidates (10 + 10
// flipped). One wave per (direction, line); lane k<20 owns one candidate.
// --------------------------------------------------------------------------
__global__ void match_lines_kernel(const float* __restrict__ lineScores,
                                   const float* __restrict__ d1, const float* __restrict__ d2,
                                   const int* __restrict__ val1, const int* __restrict__ val2,
                                   int* __restrict__ matches) {
    __shared__ float sA[S * DCH];
    __shared__ int   sAval[S];
    __shared__ int   sTopk[S];

    const int bid = blockIdx.x;
    const int dir = bid >> 9;                // 0: set1->set2, 1: set2->set1
    const int L   = bid & (NLINES - 1);
    const int lane = threadIdx.x;

    const float* dAs = dir ? d2 : d1;
    const float* dBs = dir ? d1 : d2;
    const int*   vA  = dir ? val2 : val1;
    const int*   vB  = dir ? val1 : val2;

    for (int i = lane; i < S * DCH; i += 32) sA[i] = dAs[(size_t)L * S * DCH + i];
    if (lane < S) sAval[lane] = vA[L * S + lane];
    __syncthreads();

    // ---- top-10 of this line's 512 candidate scores (argsort semantics) ----
    float lv[16];
    #pragma unroll
    for (int k = 0; k < 16; ++k) {
        int j = lane * 16 + k;
        lv[k] = dir ? lineScores[(size_t)j * NLINES + L]
                    : lineScores[(size_t)L * NLINES + j];
    }
    for (int it = 0; it < S; ++it) {
        float bv = -1e30f; int bi = 1 << 30;
        #pragma unroll
        for (int k = 0; k < 16; ++k)
            if (lv[k] > bv) { bv = lv[k]; bi = lane * 16 + k; }
        #pragma unroll
        for (int off = 16; off > 0; off >>= 1) {
            float ov = __shfl_xor(bv, off);
            int   oi = __shfl_xor(bi, off);
            if (ov > bv || (ov == bv && oi < bi)) { bv = ov; bi = oi; }
        }
        if (lane == 0) sTopk[9 - it] = bi;   // ascending order like argsort[-10:]
        #pragma unroll
        for (int k = 0; k < 16; ++k)
            if (lane * 16 + k == bi) lv[k] = -1e30f;
    }
    __syncthreads();

    // ---- NW score per candidate (lane k<20) ----
    float nw = -1e30f;
    if (lane < 20) {
        const int  cand = sTopk[lane % S];
        const bool flip = (lane >= S);
        float Smat[S][S];
        #pragma unroll
        for (int m = 0; m < S; ++m) {
            const float* brow = dBs + (size_t)(cand * S + m) * DCH;
            const int bval = vB[cand * S + m];
            float acc2[S];
            #pragma unroll
            for (int r2 = 0; r2 < S; ++r2) acc2[r2] = 0.f;
            for (int k = 0; k < DCH; ++k) {
                float b = brow[k];
                #pragma unroll
                for (int r2 = 0; r2 < S; ++r2)
                    acc2[r2] = fmaf(sA[r2 * DCH + k], b, acc2[r2]);
            }
            #pragma unroll
            for (int r2 = 0; r2 < S; ++r2) {
                float sc = (sAval[r2] && bval) ? acc2[r2] : -1.0f;
                if (flip) Smat[r2][9 - m] = sc; else Smat[r2][m] = sc;
            }
        }
        float prev[S + 1];
        #pragma unroll
        for (int j = 0; j <= S; ++j) prev[j] = 0.f;
        #pragma unroll
        for (int r2 = 0; r2 < S; ++r2) {
            float run = -1e30f;
            float pj = prev[0];
            #pragma unroll
            for (int j = 0; j < S; ++j) {
                float a = fmaxf(prev[j + 1], pj + (Smat[r2][j] - GAPC));
                run = fmaxf(run, a);
                pj = prev[j + 1];
                prev[j + 1] = fmaxf(run, 0.f);
            }
        }
        nw = prev[S];
    }
    // argmax over 20 candidates (first-occurrence tie-break)
    float bv = nw; int bi = lane;
    #pragma unroll
    for (int off = 16; off > 0; off >>= 1) {
        float ov = __shfl_xor(bv, off);
        int   oi = __shfl_xor(bi, off);
        if (ov > bv || (ov == bv && oi < bi)) { bv = ov; bi = oi; }
    }
    if (lane == 0) matches[dir * NLINES + L] = sTopk[bi % S];
}

// --------------------------------------------------------------------------
// Kernel 5: mutual check -> int32 output.
// --------------------------------------------------------------------------
__global__ void mutual_kernel(const int* __restrict__ matches, int* __restrict__ outp) {
    int i = blockIdx.x * blockDim.x + threadIdx.x;
    if (i < NLINES) {
        int m = matches[i];
        int back = matches[NLINES + m];
        outp[i] = (back == i) ? m : -1;
    }
}

// --------------------------------------------------------------------------
extern "C" void kernel_launch(void* const* d_in, const int* in_sizes, int n_in,
                              void* d_out, int out_size, void* d_ws, size_t ws_size,
                              hipStream_t stream) {
    const float* seg1  = (const float*)d_in[0];   // (512,2,2)
    const float* seg2  = (const float*)d_in[1];   // (512,2,2)
    const float* desc1 = (const float*)d_in[2];   // (1,128,128,128)
    const float* desc2 = (const float*)d_in[3];   // (1,128,128,128)

    char* ws = (char*)d_ws;
    size_t off = 0;
    float2* ptsA = (float2*)(ws + off); off += (size_t)NPTS * sizeof(float2);
    float2* ptsB = (float2*)(ws + off); off += (size_t)NPTS * sizeof(float2);
    int* valA = (int*)(ws + off); off += (size_t)NPTS * sizeof(int);
    int* valB = (int*)(ws + off); off += (size_t)NPTS * sizeof(int);
    float* d1s = (float*)(ws + off); off += (size_t)NPTS * DCH * sizeof(float);
    float* d2s = (float*)(ws + off); off += (size_t)NPTS * DCH * sizeof(float);
    float* lineScores = (float*)(ws + off); off += (size_t)NLINES * NLINES * sizeof(float);
    int* matches = (int*)(ws + off); off += 2 * NLINES * sizeof(int);

    int* outp = (int*)d_out;  // int32 output, same width as float

    sample_pts_kernel<<<4, 256, 0, stream>>>(seg1, seg2, ptsA, ptsB, valA, valB);

    grid_sample_kernel<<<2 * NPTS, DCH, 0, stream>>>(desc1, desc2, ptsA, ptsB, d1s, d2s);

    size_t smem = (size_t)160 * 161 * sizeof(float) + 320 * sizeof(int);
    gemm_linescores_kernel<<<dim3(32, 32), 256, smem, stream>>>(
        d1s, d2s, valA, valB, lineScores);

    match_lines_kernel<<<2 * NLINES, 32, 0, stream>>>(
        lineScores, d1s, d2s, valA, valB, matches);

    mutual_kernel<<<2, 256, 0, stream>>>(matches, outp);
}